// compressor_smoothing_gain_cell_9225589752163
// MI455X (gfx1250) — compile-verified
//
#include <hip/hip_runtime.h>
#include <stdint.h>

typedef uint32_t u32;
typedef uint64_t u64;
typedef u32   v4u __attribute__((ext_vector_type(4)));
typedef int   v8i __attribute__((ext_vector_type(8)));
typedef int   v4i __attribute__((ext_vector_type(4)));
typedef float vf4 __attribute__((ext_vector_type(4)));

#define LANES   32
#define CHUNK   64                 // samples per tile per row (dwords)
#define PADF    4                  // pad floats per row (TDM pad_amount = 4 dwords, keeps 16B align)
#define ROWSTR  (CHUNK + PADF)     // 68 floats
#define NBUF    6                  // LDS ring buffers
#define DEPTH   5                  // tiles prefetched ahead (NBUF-1)
#define BUFELT  (LANES * ROWSTR)   // floats per buffer

// One recurrence step: y' = x + c*(y-x), c = attack if x>=y else release.
// Chain depth: sub -> (cmp||) cndmask -> fma  (3 dependent VALU).
__device__ __forceinline__ float ar_step(float y, float x, float a, float r) {
  float d = y - x;                         // d <= 0  <=>  x >= y  (attack branch)
  float c = (d <= 0.0f) ? a : r;
  return __builtin_fmaf(c, d, x);
}

__global__ __launch_bounds__(LANES)
void compressor_cell_scan(const float* __restrict__ inp,
                          const float* __restrict__ prevs,
                          const float* __restrict__ att,
                          const float* __restrict__ rel,
                          float* __restrict__ out,
                          int B, int T) {
  __shared__ __align__(16) float sbuf[NBUF * BUFELT];

  const int lane = threadIdx.x;
  const int row0 = blockIdx.x * LANES;
  const int row  = row0 + lane;
  const bool active = (row < B);
  int rows_valid = B - row0; if (rows_valid > LANES) rows_valid = LANES;

  const float a = att[0];
  const float r = rel[0];
  float y = active ? prevs[row] : 0.0f;

  const int ntiles = T / CHUNK;
  const int Tmain  = ntiles * CHUNK;

  float* outp = out + (size_t)row * (size_t)T;

  // ---- LDS base byte offset of staging ring (low 32 bits of flat shared addr) ----
  const u32 lds_base = (u32)(u64)(uintptr_t)&sbuf[0];
  const u64 gbase    = (u64)(uintptr_t)inp + (u64)row0 * (u64)T * 4ull;

#if __has_builtin(__builtin_amdgcn_tensor_load_to_lds)
  // =====================  Tensor Data Mover staging path  =====================
  // D# group1 (constant per block): 2D tile CHUNK x 32 rows, 4B elements,
  // pad 4 dwords after every 64 dwords (row stride 68 floats in LDS).
  v8i g1;
  g1[0] = (int)((2u << 16)      // data_size = 4B
              | (1u << 20)      // pad_enable
              | (5u << 22)      // pad_interval code 5 = 64 dwords
              | (3u << 25));    // pad_amount  code 3 = 4 dwords
  g1[1] = 0;                                        // no atomic barrier; tensor_dim0 lo16 = 0
  g1[2] = (int)(0x4000u | ((u32)rows_valid << 16)); // tensor_dim0 = 1<<30 (always in-bounds); tensor_dim1 = valid rows
  g1[3] = (int)((u32)CHUNK << 16);                  // tile_dim0 = 64
  g1[4] = LANES;                                    // tile_dim1 = 32, tile_dim2 = 0
  g1[5] = (int)T;                                   // tensor_dim0_stride = T elements
  g1[6] = 0;
  g1[7] = 0;
  const v4i gz4 = {0, 0, 0, 0};
  const v8i gz8 = {0, 0, 0, 0, 0, 0, 0, 0};

  auto issue = [&](int tile) {
    u64 ga = gbase + (u64)tile * (u64)(CHUNK * 4);
    v4u g0;
    g0[0] = 1u;                                               // count = 1 (valid user D#)
    g0[1] = lds_base + (u32)(tile % NBUF) * (u32)(BUFELT * 4);// lds_addr (bytes)
    g0[2] = (u32)(ga & 0xFFFFFFFFull);                        // global_addr[31:0]
    g0[3] = (u32)((ga >> 32) & 0x01FFFFFFull) | (2u << 30);   // global_addr[56:32] | type=2
    // 6-arg form (clang-23 / therock headers): groups 2/3 + extra group all zero (2D tile).
    __builtin_amdgcn_tensor_load_to_lds(g0, g1, gz4, gz4, gz8, 0);
  };

  const int pre = (ntiles < DEPTH) ? ntiles : DEPTH;
  for (int i = 0; i < pre; ++i) issue(i);

  for (int it = 0; it < ntiles; ++it) {
    if (it + DEPTH < ntiles) {
      issue(it + DEPTH);
      __builtin_amdgcn_s_wait_tensorcnt(DEPTH - 1); // oldest tile resident (TDM ops in-order)
    } else {
      __builtin_amdgcn_s_wait_tensorcnt(0);
    }
    asm volatile("" ::: "memory"); // compiler fence: TDM wrote LDS behind its back

    const float* lrow = &sbuf[(it % NBUF) * BUFELT + lane * ROWSTR];
    if (active) {
      const int tb = it * CHUNK;
#pragma unroll
      for (int k = 0; k < CHUNK / 4; ++k) {
        vf4 xv = *(const vf4*)(lrow + 4 * k);   // ds_load_b128, off the dependent chain
        float y0 = ar_step(y,  xv.x, a, r);
        float y1 = ar_step(y0, xv.y, a, r);
        float y2 = ar_step(y1, xv.z, a, r);
        float y3 = ar_step(y2, xv.w, a, r);
        y = y3;
        outp[tb + 4 * k + 0] = y0;              // fire-and-forget global stores
        outp[tb + 4 * k + 1] = y1;
        outp[tb + 4 * k + 2] = y2;
        outp[tb + 4 * k + 3] = y3;
      }
    }
    asm volatile("" ::: "memory");
  }
#else
  // ==============  Fallback: per-lane async-to-LDS (ASYNCcnt) path  ==============
  #define DEPTH_FB 4   // 4 tiles * 16 asyncs = 64 issued, wait <= 48 (fits 6-bit imm)
  auto issue = [&](int tile) {
    u64 ga  = (u64)(uintptr_t)inp + ((u64)row * (u64)T + (u64)tile * CHUNK) * 4ull;
    u32 lad = lds_base + (u32)(tile % NBUF) * (u32)(BUFELT * 4) + (u32)lane * (ROWSTR * 4);
#pragma unroll
    for (int k = 0; k < CHUNK / 4; ++k) {
      asm volatile("global_load_async_to_lds_b128 %0, %1, off"
                   :: "v"(lad + 16u * k), "v"(ga + 16ull * k) : "memory");
    }
  };
  const int pre = (ntiles < DEPTH_FB) ? ntiles : DEPTH_FB;
  for (int i = 0; i < pre; ++i) issue(i);

  for (int it = 0; it < ntiles; ++it) {
    if (it + DEPTH_FB < ntiles) {
      issue(it + DEPTH_FB);
      asm volatile("s_wait_asynccnt 48" ::: "memory");
    } else {
      asm volatile("s_wait_asynccnt 0" ::: "memory");
    }
    const float* lrow = &sbuf[(it % NBUF) * BUFELT + lane * ROWSTR];
    if (active) {
      const int tb = it * CHUNK;
#pragma unroll
      for (int k = 0; k < CHUNK / 4; ++k) {
        vf4 xv = *(const vf4*)(lrow + 4 * k);
        float y0 = ar_step(y,  xv.x, a, r);
        float y1 = ar_step(y0, xv.y, a, r);
        float y2 = ar_step(y1, xv.z, a, r);
        float y3 = ar_step(y2, xv.w, a, r);
        y = y3;
        outp[tb + 4 * k + 0] = y0;
        outp[tb + 4 * k + 1] = y1;
        outp[tb + 4 * k + 2] = y2;
        outp[tb + 4 * k + 3] = y3;
      }
    }
    asm volatile("" ::: "memory");
  }
#endif

  // ---- tail (T % CHUNK != 0); direct global loads ----
  if (active) {
    for (int t = Tmain; t < T; ++t) {
      float x = inp[(size_t)row * (size_t)T + t];
      y = ar_step(y, x, a, r);
      outp[t] = y;
    }
  }
}

extern "C" void kernel_launch(void* const* d_in, const int* in_sizes, int n_in,
                              void* d_out, int out_size, void* d_ws, size_t ws_size,
                              hipStream_t stream) {
  (void)n_in; (void)out_size; (void)d_ws; (void)ws_size;
  const float* inp  = (const float*)d_in[0];  // [B, T]
  const float* prev = (const float*)d_in[1];  // [B]
  const float* att  = (const float*)d_in[2];  // scalar
  const float* rel  = (const float*)d_in[3];  // scalar
  float* out = (float*)d_out;                 // [B, T]

  const int B = in_sizes[1];
  const int T = in_sizes[0] / B;
  const int grid = (B + LANES - 1) / LANES;

  hipLaunchKernelGGL(compressor_cell_scan, dim3(grid), dim3(LANES), 0, stream,
                     inp, prev, att, rel, out, B, T);
}